// ConstraintsModule_49778670960855
// MI455X (gfx1250) — compile-verified
//
#include <hip/hip_runtime.h>
#include <stdint.h>

#define BATCH 16384
#define NUM   512
#define CONS  512
#define ROWS  128          // batch rows per workgroup == blockDim.x
#define RS    (ROWS + 1)   // padded LDS row-stride (odd -> conflict-free in both phases)

// ---- CDNA5 (gfx1250) async global<->LDS DMA helpers ------------------------
// VGLOBAL async ops: LDS byte offset in a VGPR, 64-bit global address in a VGPR
// pair, tracked with ASYNCcnt (CDNA5 ISA).
__device__ __forceinline__ void async_load_to_lds_b32(uint32_t lds_byte_off,
                                                      const void* gaddr) {
  asm volatile("global_load_async_to_lds_b32 %0, %1, off"
               :: "v"(lds_byte_off), "v"(gaddr)
               : "memory");
}
__device__ __forceinline__ void async_store_from_lds_b32(void* gaddr,
                                                         uint32_t lds_byte_off) {
  asm volatile("global_store_async_from_lds_b32 %0, %1, off"
               :: "v"(gaddr), "v"(lds_byte_off)
               : "memory");
}
__device__ __forceinline__ void wait_asynccnt0() {
  asm volatile("s_wait_asynccnt 0" ::: "memory");
}

// ---- pass 1: compact the disjoint binary body masks into sparse literal lists
// entry = atom index | (0x80000000 if negative literal). ~85 entries/constraint.
__global__ void build_sparse_kernel(const float* __restrict__ pos,
                                    const float* __restrict__ neg,
                                    uint32_t* __restrict__ ent,
                                    int* __restrict__ cnt) {
  const int c = blockIdx.x * blockDim.x + threadIdx.x;
  if (c >= CONS) return;
  const float* pr = pos + (size_t)c * NUM;
  const float* nr = neg + (size_t)c * NUM;
  uint32_t* er = ent + (size_t)c * NUM;
  int n = 0;
  for (int j = 0; j < NUM; ++j) {
    const float pv = pr[j];
    const float nv = nr[j];
    if (pv != 0.0f) {
      er[n++] = (uint32_t)j;                 // positive literal: val = 1 - p = |p - 1|
    } else if (nv != 0.0f) {
      er[n++] = (uint32_t)j | 0x80000000u;   // negative literal: val = p     = |p - 0|
    }
  }
  cnt[c] = n;
}

// ---- pass 2: Gauss-Seidel sweep, one batch row per lane, p-tile LDS-resident
extern __shared__ float ldsp[];              // NUM * RS floats  (~258 KB of 320 KB WGP LDS)

__device__ __forceinline__ float body_val(uint32_t ev, int t) {
  const int j = (int)(ev & 0xffffu);
  const float sel = (ev & 0x80000000u) ? 0.0f : 1.0f;  // uniform scalar select
  const float pv = ldsp[j * RS + t];                   // ds_load, lanes consecutive
  return fabsf(pv - sel);                              // pos: 1-p ; neg: p
}

__global__ __launch_bounds__(ROWS, 1)
void gauss_seidel_scan_kernel(const float* __restrict__ preds,
                              const uint32_t* __restrict__ ent,
                              const int* __restrict__ cnt,
                              const int* __restrict__ head_atom,
                              const unsigned char* __restrict__ head_sign,
                              float* __restrict__ out) {
  const int t = threadIdx.x;                 // lane owns batch row rowBase + t
  const int rowBase = blockIdx.x * ROWS;
  // Generic LDS pointers are {shared_aperture[63:32], lds_offset[31:0]}:
  // low 32 bits are exactly the wave-relative LDS byte address the DMA wants.
  const uint32_t lds_base = (uint32_t)(uintptr_t)&ldsp[0];

  // Stage p tile into LDS, transposed (lds[j*RS + r] = p[rowBase+r][j]) via
  // async DMA: coalesced global reads, per-lane LDS destinations do the
  // transpose, RS=129 keeps every phase bank-conflict-free.
  for (int r = 0; r < ROWS; ++r) {
    const float* gp = preds + (size_t)(rowBase + r) * NUM;
#pragma unroll
    for (int ch = 0; ch < NUM / ROWS; ++ch) {
      const int j = ch * ROWS + t;
      async_load_to_lds_b32(lds_base + (uint32_t)((j * RS + r) * sizeof(float)),
                            gp + j);
    }
  }
  wait_asynccnt0();
  __syncthreads();

  // Sequential sweep over constraints. All control/metadata is wave-uniform
  // (scalar loads); only the gather + max chain is per-lane VALU/DS work.
  // Each lane exclusively owns its row's LDS slots -> no barriers in the loop.
  for (int c = 0; c < CONS; ++c) {
    const int n = cnt[c];
    const uint32_t* e = ent + (size_t)c * NUM;
    // 4-wide unroll: one 16B uniform entry fetch + 4 independent ds_loads and
    // max accumulators per iteration (breaks the serial load->max chain).
    float m0 = 0.0f, m1 = 0.0f, m2 = 0.0f, m3 = 0.0f;
    int k = 0;
    for (; k + 4 <= n; k += 4) {
      const uint4 ev = *reinterpret_cast<const uint4*>(e + k);  // s_load_b128
      m0 = fmaxf(m0, body_val(ev.x, t));
      m1 = fmaxf(m1, body_val(ev.y, t));
      m2 = fmaxf(m2, body_val(ev.z, t));
      m3 = fmaxf(m3, body_val(ev.w, t));
    }
    for (; k < n; ++k) {
      m0 = fmaxf(m0, body_val(e[k], t));
    }
    const float m = fmaxf(fmaxf(m0, m1), fmaxf(m2, m3));  // = max(0, max_j val)
    // cand = 1 - m;  positive head: max(prev, cand); negative: min(prev, 1-cand)=min(prev, m)
    const int h = head_atom[c];
    const float prev = ldsp[h * RS + t];
    const float upd = head_sign[c] ? fmaxf(prev, 1.0f - m)
                                   : fminf(prev, m);
    ldsp[h * RS + t] = upd;
  }
  __syncthreads();

  // Write the final tile back with coalesced async LDS->global DMA.
  for (int r = 0; r < ROWS; ++r) {
    float* gd = out + (size_t)(rowBase + r) * NUM;
#pragma unroll
    for (int ch = 0; ch < NUM / ROWS; ++ch) {
      const int j = ch * ROWS + t;
      async_store_from_lds_b32(gd + j,
                               lds_base + (uint32_t)((j * RS + r) * sizeof(float)));
    }
  }
  wait_asynccnt0();   // S_ENDPGM also implies wait-idle, but be explicit
}

extern "C" void kernel_launch(void* const* d_in, const int* in_sizes, int n_in,
                              void* d_out, int out_size, void* d_ws, size_t ws_size,
                              hipStream_t stream) {
  const float* preds        = (const float*)d_in[0];
  const float* pos_body     = (const float*)d_in[1];
  const float* neg_body     = (const float*)d_in[2];
  const int*   head_atom    = (const int*)d_in[3];
  const unsigned char* sign = (const unsigned char*)d_in[4];  // jnp.bool_ -> 1 byte
  float* out = (float*)d_out;

  // Workspace layout: [CONS*NUM u32 entries][CONS i32 counts]  (~1.05 MB)
  uint32_t* ent = (uint32_t*)d_ws;
  int* cnt = (int*)((char*)d_ws + (size_t)CONS * NUM * sizeof(uint32_t));

  build_sparse_kernel<<<(CONS + 255) / 256, 256, 0, stream>>>(
      pos_body, neg_body, ent, cnt);

  const int batch = in_sizes[0] / NUM;
  const size_t ldsBytes = (size_t)NUM * RS * sizeof(float);   // 264192 B
  (void)hipFuncSetAttribute((const void*)gauss_seidel_scan_kernel,
                            hipFuncAttributeMaxDynamicSharedMemorySize,
                            (int)ldsBytes);
  gauss_seidel_scan_kernel<<<batch / ROWS, ROWS, ldsBytes, stream>>>(
      preds, ent, cnt, head_atom, sign, out);
}